// SAGEConv_33509334843591
// MI455X (gfx1250) — compile-verified
//
#include <hip/hip_runtime.h>

typedef float v2f __attribute__((ext_vector_type(2)));
typedef float v8f __attribute__((ext_vector_type(8)));

#define DFEAT  128      // D_IN == D_OUT
#define KTOT   256      // concatenated K: [mean | x]
#define ROWS   128      // rows per block in GEMM kernel
#define KC     32       // K-chunk staged in LDS
#define NCHUNK (KTOT / KC)

// Low 32 bits of a generic pointer to LDS == the DS byte address.
__device__ __forceinline__ unsigned lds_addr(const void* p) {
    return (unsigned)(size_t)p;
}

// Async 16B global -> LDS copy (GVS mode), tracked by ASYNCcnt.
__device__ __forceinline__ void async_copy16(unsigned lds, unsigned voff,
                                             const void* sbase) {
    asm volatile("global_load_async_to_lds_b128 %0, %1, %2"
                 :: "v"(lds), "v"(voff), "s"(sbase)
                 : "memory");
}
__device__ __forceinline__ void wait_async0() {
    asm volatile("s_wait_asynccnt 0x0" ::: "memory");
}

// ---------------------------------------------------------------------------
// Kernel 1: zero the scatter accumulators (msg_sum + cnt) in workspace
// ---------------------------------------------------------------------------
__global__ void sage_zero_ws(float* __restrict__ ws, long long total) {
    long long i4 = ((long long)blockIdx.x * blockDim.x + threadIdx.x) * 4;
    if (i4 + 3 < total) {
        *(float4*)(ws + i4) = make_float4(0.f, 0.f, 0.f, 0.f);
    } else {
        for (long long i = i4; i < total; ++i) ws[i] = 0.f;
    }
}

// ---------------------------------------------------------------------------
// Kernel 2: build permuted WTp:  WTp[(k>>1)*256 + j*2 + (k&1)] = Wcat^T[k][j]
// (k-pair interleave so a B fragment is a single aligned ds_load_b64)
// ---------------------------------------------------------------------------
__global__ void sage_build_wt(const float* __restrict__ Wrel,
                              const float* __restrict__ Wroot,
                              float* __restrict__ WTp) {
    int g = blockIdx.x * blockDim.x + threadIdx.x;   // 0 .. 32767
    int k = g >> 7;
    int j = g & 127;
    float v = (k < DFEAT) ? Wrel[j * DFEAT + k] : Wroot[j * DFEAT + (k - DFEAT)];
    WTp[(k >> 1) * 256 + (j << 1) + (k & 1)] = v;
}

// ---------------------------------------------------------------------------
// Kernel 3: scatter-add. One wave (32 lanes) per edge; each lane moves 4 floats.
// msg_sum[col] += x[row]; cnt[col] += 1   (accumulators stay L2-resident)
// ---------------------------------------------------------------------------
__global__ void sage_scatter(const float* __restrict__ x,
                             const int* __restrict__ ei,   // [2, E] flat
                             float* __restrict__ msg,
                             float* __restrict__ cnt,
                             int E) {
    long long t = (long long)blockIdx.x * blockDim.x + threadIdx.x;
    int e    = (int)(t >> 5);
    int lane = (int)(t & 31);
    if (e >= E) return;
    int r = ei[e];
    int c = ei[E + e];
    float4 v = ((const float4*)(x + (size_t)r * DFEAT))[lane];
    float* m = msg + (size_t)c * DFEAT + lane * 4;
    atomicAdd(m + 0, v.x);
    atomicAdd(m + 1, v.y);
    atomicAdd(m + 2, v.z);
    atomicAdd(m + 3, v.w);
    if (lane == 0) atomicAdd(cnt + c, 1.0f);
}

// ---------------------------------------------------------------------------
// Kernel 4: fused  out = normalize( [msg/cnt | x] @ Wcat^T + b )
// 256 threads = 8 waves; block covers 128 rows; each wave owns 16 rows x 128 cols.
// FP32 WMMA 16x16x4. K streamed via double-buffered async global->LDS copies.
// kc loop fully unrolled so the mean-scale and source selection are static.
// ---------------------------------------------------------------------------
__global__ void __launch_bounds__(256)
sage_gemm_norm(const float* __restrict__ x,
               const float* __restrict__ msg,
               const float* __restrict__ cnt,
               const float* __restrict__ WTp,    // permuted [128][128][2]
               const float* __restrict__ bias,   // [128]
               float* __restrict__ out,
               int N) {
    __shared__ float sA[2][ROWS * KC];      // 2 x 16 KB, [row][kk]
    __shared__ float sB[2][KC * DFEAT];     // 2 x 16 KB, [kk/2][j][kk&1]

    const int tid  = threadIdx.x;
    const int wave = tid >> 5;
    const int lane = tid & 31;
    const int half = lane >> 4;          // 0: lanes 0-15, 1: lanes 16-31
    const int n16  = lane & 15;
    const int base = blockIdx.x * ROWS;
    const int arow = wave * 16 + n16;    // A fragment row for this lane
    const int koff = half * 2;           // upper half-wave holds K+2, K+3

    // per-lane constants (no LDS needed for these)
    int growA = base + arow; if (growA >= N) growA = N - 1;
    const float inv_a = 1.0f / fmaxf(cnt[growA], 1.0f);
    float bs[8];
#pragma unroll
    for (int nt = 0; nt < 8; ++nt) bs[nt] = bias[nt * 16 + n16];

    // -------- async staging of one K-chunk into buffer p ------------------
    auto issue = [&](int kc, int p) {
        // A chunk: 128 rows x 32 k = 16 KB; 4 x b128 per thread.
        // Chunk kc<4 comes from msg (scaled later in-register), else from x.
        const char* srcA = (kc < 4)
            ? ((const char*)msg + (size_t)kc * 128)
            : ((const char*)x + (size_t)(kc - 4) * 128);
        unsigned ldsA = lds_addr(&sA[p][0]);
#pragma unroll
        for (int i = 0; i < 4; ++i) {
            int idx4 = i * 256 + tid;            // 0..1023 float4 slots
            int row  = idx4 >> 3;                // 0..127
            int kk4  = idx4 & 7;                 // 0..7
            int grow = base + row; if (grow >= N) grow = N - 1;  // clamp, stores guarded
            unsigned voff = (unsigned)grow * 512u + (unsigned)kk4 * 16u;
            async_copy16(ldsA + (unsigned)(row * 128 + kk4 * 16), voff, srcA);
        }
        // B chunk: contiguous 16 KB slice of permuted WTp.
        const char* srcB = (const char*)WTp + (size_t)kc * 16384;
        unsigned ldsB = lds_addr(&sB[p][0]);
#pragma unroll
        for (int i = 0; i < 4; ++i) {
            unsigned voff = (unsigned)((i * 256 + tid) * 16);
            async_copy16(ldsB + voff, voff, srcB);
        }
    };

    issue(0, 0);

    v8f acc[8];
#pragma unroll
    for (int nt = 0; nt < 8; ++nt)
#pragma unroll
        for (int r = 0; r < 8; ++r) acc[nt][r] = 0.f;

#pragma unroll
    for (int kc = 0; kc < NCHUNK; ++kc) {
        const int p = kc & 1;
        wait_async0();          // my copies for chunk kc are in LDS
        __syncthreads();        // everyone's copies visible; prev reads done
        if (kc + 1 < NCHUNK) issue(kc + 1, p ^ 1);

        const bool  scale = (kc < 4);        // compile-time after full unroll
        const float* Abuf = sA[p];
        const float* Bbuf = sB[p];
#pragma unroll
        for (int kb = 0; kb < KC; kb += 4) {
            v2f a = *(const v2f*)&Abuf[arow * KC + kb + koff];
            if (scale) { a.x *= inv_a; a.y *= inv_a; }
#pragma unroll
            for (int nt = 0; nt < 8; ++nt) {
                v2f b = *(const v2f*)&Bbuf[((kb + koff) >> 1) * 256 + (nt * 16 + n16) * 2];
                acc[nt] = __builtin_amdgcn_wmma_f32_16x16x4_f32(
                    /*neg_a=*/false, a, /*neg_b=*/false, b,
                    /*c_mod=*/(short)0, acc[nt],
                    /*reuse_a=*/false, /*reuse_b=*/false);
            }
        }
    }

    // ---- bias + row L2 norm + store
    // C/D layout: VGPR r holds element (M = r + 8*half, N = n16) of tile nt.
    float ss[8];
#pragma unroll
    for (int r = 0; r < 8; ++r) ss[r] = 0.f;
#pragma unroll
    for (int nt = 0; nt < 8; ++nt)
#pragma unroll
        for (int r = 0; r < 8; ++r) {
            float v = acc[nt][r] + bs[nt];
            acc[nt][r] = v;
            ss[r] += v * v;
        }
#pragma unroll
    for (int r = 0; r < 8; ++r) {
        float s = ss[r];
        s += __shfl_xor(s, 1, 32);
        s += __shfl_xor(s, 2, 32);
        s += __shfl_xor(s, 4, 32);
        s += __shfl_xor(s, 8, 32);
        ss[r] = 1.0f / fmaxf(sqrtf(s), 1e-12f);
    }
#pragma unroll
    for (int r = 0; r < 8; ++r) {
        int grow = base + wave * 16 + r + half * 8;
        if (grow < N) {
            float inv = ss[r];
#pragma unroll
            for (int nt = 0; nt < 8; ++nt)
                __builtin_nontemporal_store(acc[nt][r] * inv,
                    &out[(size_t)grow * DFEAT + nt * 16 + n16]);
        }
    }
}

// ---------------------------------------------------------------------------
extern "C" void kernel_launch(void* const* d_in, const int* in_sizes, int n_in,
                              void* d_out, int out_size, void* d_ws, size_t ws_size,
                              hipStream_t stream) {
    const float* x     = (const float*)d_in[0];   // [N,128]
    const int*   ei    = (const int*)d_in[2];     // [2,E]
    const float* Wrel  = (const float*)d_in[3];   // [128,128]
    const float* brel  = (const float*)d_in[4];   // [128]
    const float* Wroot = (const float*)d_in[5];   // [128,128]
    float*       out   = (float*)d_out;

    const int N = in_sizes[0] / DFEAT;
    const int E = in_sizes[2] / 2;

    float* msg = (float*)d_ws;                     // N*128
    float* cnt = msg + (size_t)N * DFEAT;          // N
    float* WTp = cnt + N;                          // 256*128 (permuted)

    // 1) zero accumulators
    long long ztot = (long long)N * DFEAT + N;
    long long zthreads = (ztot + 3) / 4;
    sage_zero_ws<<<(unsigned)((zthreads + 255) / 256), 256, 0, stream>>>(msg, ztot);

    // 2) weight transpose/concat/permute (32768 elements)
    sage_build_wt<<<(KTOT * DFEAT) / 256, 256, 0, stream>>>(Wrel, Wroot, WTp);

    // 3) scatter-mean accumulation (one wave per edge)
    long long sthreads = (long long)E * 32;
    sage_scatter<<<(unsigned)((sthreads + 255) / 256), 256, 0, stream>>>(x, ei, msg, cnt, E);

    // 4) fused GEMM + bias + L2 normalize (double-buffered async staging)
    int gblocks = (N + ROWS - 1) / ROWS;
    sage_gemm_norm<<<gblocks, 256, 0, stream>>>(x, msg, cnt, WTp, brel, out, N);
}